// yoloLoss_4363686773032
// MI455X (gfx1250) — compile-verified
//
#include <hip/hip_runtime.h>

// ---------------------------------------------------------------------------
// YOLO loss, MI455X (gfx1250, wave32).
// Streaming map-reduce: per-cell loss in registers from coalesced b128 loads,
// wave reduction via V_WMMA_F32_16X16X4_F32 ones-matrix trick (5 WMMAs reduce
// all 32 lanes), block reduce in LDS, deterministic 2-stage finalize.
// Round 2: finalize kernel parallelized (256 threads + same WMMA reduction)
// so the partial-sum tail no longer serializes ~1568 dependent adds.
// ---------------------------------------------------------------------------

typedef float v2f __attribute__((ext_vector_type(2)));
typedef float v8f __attribute__((ext_vector_type(8)));

#define BLOCK 256
#define MAX_PARTIALS 8192

__device__ __forceinline__ float cell_loss(const float* __restrict__ p,
                                           const float* __restrict__ t) {
    const float EPS = 1e-6f;
    bool is_obj = t[4] > 0.0f;

    // no-object confidence loss (channels 4 and 9)
    float d4 = p[4] - t[4];
    float d9 = p[9] - t[9];
    float noobj = d4 * d4 + d9 * d9;

    // class loss (single class, channel 10)
    float dcl = p[10] - t[10];
    float cls = dcl * dcl;

    // target box 0 in xyxy
    float t_x1 = t[0] - 0.5f * t[2];
    float t_y1 = t[1] - 0.5f * t[3];
    float t_x2 = t[0] + 0.5f * t[2];
    float t_y2 = t[1] + 0.5f * t[3];
    float a2 = (t_x2 - t_x1) * (t_y2 - t_y1);

    // IoU of both predicted boxes vs target box 0 (unclamped inter, as in ref)
    float iou0, iou1;
    {
        float x1 = p[0] - 0.5f * p[2], y1 = p[1] - 0.5f * p[3];
        float x2 = p[0] + 0.5f * p[2], y2 = p[1] + 0.5f * p[3];
        float inter = (fminf(x2, t_x2) - fmaxf(x1, t_x1)) *
                      (fminf(y2, t_y2) - fmaxf(y1, t_y1));
        float a1 = (x2 - x1) * (y2 - y1);
        iou0 = inter / (a1 + a2 - inter);
    }
    {
        float x1 = p[5] - 0.5f * p[7], y1 = p[6] - 0.5f * p[8];
        float x2 = p[5] + 0.5f * p[7], y2 = p[6] + 0.5f * p[8];
        float inter = (fminf(x2, t_x2) - fmaxf(x1, t_x1)) *
                      (fminf(y2, t_y2) - fmaxf(y1, t_y1));
        float a1 = (x2 - x1) * (y2 - y1);
        iou1 = inter / (a1 + a2 - inter);
    }

    bool r1 = iou1 > iou0;                 // argmax (ties -> box 0, like jnp)
    float miou = fmaxf(iou0, iou1);

    // responsible pred/target box via register selects (stay out of scratch)
    float prx = r1 ? p[5] : p[0];
    float pry = r1 ? p[6] : p[1];
    float prw = r1 ? p[7] : p[2];
    float prh = r1 ? p[8] : p[3];
    float prc = r1 ? p[9] : p[4];
    float trx = r1 ? t[5] : t[0];
    float try_ = r1 ? t[6] : t[1];
    float trw = r1 ? t[7] : t[2];
    float trh = r1 ? t[8] : t[3];

    float dcf = prc - miou;
    float contain = dcf * dcf;

    float dx = prx - trx;
    float dy = pry - try_;
    float dw = sqrtf(prw + EPS) - sqrtf(trw + EPS);
    float dh = sqrtf(prh + EPS) - sqrtf(trh + EPS);
    float loc = dx * dx + dy * dy + dw * dw + dh * dh;

    float pos = 5.0f * loc + contain + cls;     // L_COORD = 5
    return (is_obj ? pos : 0.5f * noobj);        // L_NOOBJ = 0.5
}

// Sum `x` across all 32 lanes using V_WMMA_F32_16X16X4_F32 with ones-B.
// Stage 1: A = lane partials (K0/K2 slots), B = ones -> D1[m][n] = x_m + x_{m+16}.
// Stage 2: 4 accumulating WMMAs feed all 8 D1 VGPRs back as A-matrices; each
// row of the combined A2s covers all 16 D1 rows at a fixed column, so
// D2[m][n] = sum over all 32 lanes. EXEC must be all 1s here (called only
// after full reconvergence with a 256-thread block).
__device__ __forceinline__ float wave_reduce_wmma(float x) {
    v2f a;    a[0] = x;       a[1] = 0.0f;
    v2f ones; ones[0] = 1.0f; ones[1] = 1.0f;
    v8f z = {};
    v8f d1 = __builtin_amdgcn_wmma_f32_16x16x4_f32(
        false, a, false, ones, (short)0, z, false, false);
    v8f acc = {};
#pragma unroll
    for (int j = 0; j < 8; j += 2) {
        v2f aj; aj[0] = d1[j]; aj[1] = d1[j + 1];
        acc = __builtin_amdgcn_wmma_f32_16x16x4_f32(
            false, aj, false, ones, (short)0, acc, false, false);
    }
    return acc[0];
}

// Block-level reduction: WMMA wave reduce + LDS combine; returns full block
// sum in thread 0 (valid only there).
__device__ __forceinline__ float block_reduce(float part) {
    float wsum = wave_reduce_wmma(part);
    __shared__ float smem[BLOCK / 32];
    int lane = threadIdx.x & 31;
    int wid  = threadIdx.x >> 5;
    if (lane == 0) smem[wid] = wsum;
    __syncthreads();
    float s = 0.0f;
    if (threadIdx.x == 0) {
#pragma unroll
        for (int i = 0; i < BLOCK / 32; ++i) s += smem[i];
    }
    return s;
}

__global__ void __launch_bounds__(BLOCK)
yolo_main(const float* __restrict__ pred, const float* __restrict__ tgt,
          long long ngroups, float* __restrict__ partials) {
    long long tid = (long long)blockIdx.x * blockDim.x + threadIdx.x;
    long long stride = (long long)gridDim.x * blockDim.x;

    float part = 0.0f;
    // One group = 4 consecutive cells = 44 floats = 11 float4 (16B aligned).
    for (long long g = tid; g < ngroups; g += stride) {
        const float4* __restrict__ pg = reinterpret_cast<const float4*>(pred) + g * 11;
        const float4* __restrict__ tg = reinterpret_cast<const float4*>(tgt) + g * 11;
        float P[44], T[44];
#pragma unroll
        for (int i = 0; i < 11; ++i) {
            float4 v = pg[i];
            P[4 * i + 0] = v.x; P[4 * i + 1] = v.y;
            P[4 * i + 2] = v.z; P[4 * i + 3] = v.w;
            float4 u = tg[i];
            T[4 * i + 0] = u.x; T[4 * i + 1] = u.y;
            T[4 * i + 2] = u.z; T[4 * i + 3] = u.w;
        }
#pragma unroll
        for (int c = 0; c < 4; ++c)
            part += cell_loss(&P[c * 11], &T[c * 11]);
    }

    float s = block_reduce(part);
    if (threadIdx.x == 0) partials[blockIdx.x] = s;
}

// Deterministic finalize: 256 threads grid-stride over block partials, same
// WMMA+LDS reduction, thread 0 adds any tail cells, scales by 1/Nb and
// overwrites d_out (harness poisons it).
__global__ void __launch_bounds__(BLOCK)
yolo_finalize(const float* __restrict__ partials, int nblocks,
              const float* __restrict__ pred,
              const float* __restrict__ tgt,
              long long tail_start, long long n_cells,
              float inv_nb, float* __restrict__ out) {
    float part = 0.0f;
    for (int i = threadIdx.x; i < nblocks; i += BLOCK) part += partials[i];

    float s = block_reduce(part);

    if (threadIdx.x == 0) {
        for (long long c = tail_start; c < n_cells; ++c) {
            float p[11], t[11];
#pragma unroll
            for (int k = 0; k < 11; ++k) {
                p[k] = pred[c * 11 + k];
                t[k] = tgt[c * 11 + k];
            }
            s += cell_loss(p, t);
        }
        out[0] = s * inv_nb;
    }
}

extern "C" void kernel_launch(void* const* d_in, const int* in_sizes, int n_in,
                              void* d_out, int out_size, void* d_ws, size_t ws_size,
                              hipStream_t stream) {
    const float* pred = (const float*)d_in[0];
    const float* tgt  = (const float*)d_in[1];
    float* out = (float*)d_out;
    float* partials = (float*)d_ws;

    long long n_elem = in_sizes[0];                 // Nb*S*S*11
    long long n_cells = n_elem / 11;
    long long ngroups = n_cells / 4;
    long long tail_start = ngroups * 4;
    float inv_nb = 1.0f / (float)(n_elem / (28LL * 28LL * 11LL));  // Nb per ref shape

    int want = (int)((ngroups + BLOCK - 1) / BLOCK);               // 1568 for ref
    int cap = (int)(ws_size / sizeof(float));
    if (cap > MAX_PARTIALS) cap = MAX_PARTIALS;
    int nblocks = want < cap ? want : cap;
    if (nblocks < 1) nblocks = 1;

    yolo_main<<<nblocks, BLOCK, 0, stream>>>(pred, tgt, ngroups, partials);
    yolo_finalize<<<1, BLOCK, 0, stream>>>(partials, nblocks, pred, tgt,
                                           tail_start, n_cells, inv_nb, out);
}